// VolumeSDFRenderer_6012954214895
// MI455X (gfx1250) — compile-verified
//
#include <hip/hip_runtime.h>
#include <hip/hip_bf16.h>

#define ALPHA      10.0f
#define HALF_ALPHA 5.0f
#define INV_BETA   20.0f      // 1 / 0.05
#define FAR_DELTA  1e10f

// ---- CDNA5 async global->LDS path (probe via __has_builtin; fallback = plain b128 loads)
#if defined(__gfx1250__) && __has_builtin(__builtin_amdgcn_global_load_async_to_lds_b128) && __has_builtin(__builtin_amdgcn_s_wait_asynccnt)
#define USE_ASYNC_LDS 1
#else
#define USE_ASYNC_LDS 0
#endif

#if USE_ASYNC_LDS
// Builtin signature (from hipcc diagnostic): param0 = AS1 (global) int4*, param1 = AS3 (LDS) int4*,
// then two integer immediates (offset, cpol).
typedef int vi4 __attribute__((vector_size(16)));
typedef __attribute__((address_space(1))) vi4* g4p;
typedef __attribute__((address_space(3))) vi4* l4p;
#endif

__device__ __forceinline__ float sdf_density(float dist) {
    // s = -dist; s<=0: alpha*0.5*exp(s/beta); s>0: alpha*(1 - 0.5*exp(-s/beta))
    float s = -dist;
    float e = expf(-fabsf(s) * INV_BETA);            // == exp(s/beta) for s<=0, exp(-s/beta) for s>0
    return (s <= 0.0f) ? HALF_ALPHA * e : (ALPHA - HALF_ALPHA * e);
}

// One wave (32 lanes) per ray; lane owns samples [4*lane, 4*lane+4).
__global__ __launch_bounds__(256) void render_rays(
    const float* __restrict__ distance,   // [R,128]
    const float* __restrict__ color,      // [R,128,3]
    const float* __restrict__ depth,      // [R,128]
    float* __restrict__ out_color,        // [R,3]
    int R)
{
    const int lane = threadIdx.x & 31;
    const int wave = threadIdx.x >> 5;
    const int ray  = blockIdx.x * 8 + wave;
    if (ray >= R) return;                 // wave-uniform

    const float4* c4 = (const float4*)(color + (size_t)ray * 384);
    __builtin_prefetch(c4 + lane * 3, 0, 1);   // global_prefetch_b8: warm color while depth/dist land

    float4 dv, tv;
#if USE_ASYNC_LDS
    __shared__ float stage[8][256];       // per wave: [0,128) distance, [128,256) depth
    {
        const float* gd = distance + (size_t)ray * 128 + lane * 4;
        const float* gt = depth    + (size_t)ray * 128 + lane * 4;
        __builtin_amdgcn_global_load_async_to_lds_b128(
            (g4p)gd, (l4p)&stage[wave][lane * 4], 0, 0);
        __builtin_amdgcn_global_load_async_to_lds_b128(
            (g4p)gt, (l4p)&stage[wave][128 + lane * 4], 0, 0);
        __builtin_amdgcn_s_wait_asynccnt(0);
        asm volatile("" ::: "memory");    // compiler-level ordering for LDS reads below
        dv = *(const float4*)&stage[wave][lane * 4];
        tv = *(const float4*)&stage[wave][128 + lane * 4];
    }
#else
    dv = *(const float4*)(distance + (size_t)ray * 128 + lane * 4);
    tv = *(const float4*)(depth    + (size_t)ray * 128 + lane * 4);
#endif

    // deltas: delta[n] = depth[n+1]-depth[n], delta[127] = FAR_DELTA
    float tnext = __shfl_down(tv.x, 1, 32);          // depth[4*(lane+1)] == depth[4*lane+4]
    float de0 = tv.y - tv.x;
    float de1 = tv.z - tv.y;
    float de2 = tv.w - tv.z;
    float de3 = (lane == 31) ? FAR_DELTA : (tnext - tv.w);

    float dd0 = sdf_density(dv.x) * de0;
    float dd1 = sdf_density(dv.y) * de1;
    float dd2 = sdf_density(dv.z) * de2;
    float dd3 = sdf_density(dv.w) * de3;

    // exclusive scan over the ray: lane-local prefixes + wave32 shfl_up scan of lane sums
    float l1 = dd0, l2 = dd0 + dd1, l3 = l2 + dd2;
    float laneSum = l3 + dd3;
    float x = laneSum;
    #pragma unroll
    for (int off = 1; off < 32; off <<= 1) {
        float y = __shfl_up(x, off, 32);
        if (lane >= off) x += y;
    }
    float excl = x - laneSum;                        // exclusive prefix of lane sums

    float c0 = excl, c1 = excl + l1, c2 = excl + l2, c3 = excl + l3;
    float w0 = (1.0f - expf(-dd0)) * expf(-c0);
    float w1 = (1.0f - expf(-dd1)) * expf(-c1);
    float w2 = (1.0f - expf(-dd2)) * expf(-c2);
    float w3 = (1.0f - expf(-dd3)) * expf(-c3);

    // 48B of contiguous color per lane (3x float4), fully coalesced per wave
    float4 ca = c4[lane * 3 + 0];   // n0.rgb, n1.r
    float4 cb = c4[lane * 3 + 1];   // n1.gb, n2.rg
    float4 cc = c4[lane * 3 + 2];   // n2.b, n3.rgb

    float ar = w0 * ca.x + w1 * ca.w + w2 * cb.z + w3 * cc.y;
    float ag = w0 * ca.y + w1 * cb.x + w2 * cb.w + w3 * cc.z;
    float ab = w0 * ca.z + w1 * cb.y + w2 * cc.x + w3 * cc.w;

    #pragma unroll
    for (int off = 16; off >= 1; off >>= 1) {
        ar += __shfl_xor(ar, off, 32);
        ag += __shfl_xor(ag, off, 32);
        ab += __shfl_xor(ab, off, 32);
    }
    if (lane == 0) {
        out_color[(size_t)ray * 3 + 0] = ar;
        out_color[(size_t)ray * 3 + 1] = ag;
        out_color[(size_t)ray * 3 + 2] = ab;
    }
}

// geometry output = zeros_like(color): pure streaming b128 stores (dominant traffic, ~100 MB)
__global__ __launch_bounds__(256) void zero_geometry(float4* __restrict__ p, size_t n4) {
    size_t i      = (size_t)blockIdx.x * blockDim.x + threadIdx.x;
    size_t stride = (size_t)gridDim.x * blockDim.x;
    const float4 z = make_float4(0.f, 0.f, 0.f, 0.f);
    for (; i < n4; i += stride) p[i] = z;
}

extern "C" void kernel_launch(void* const* d_in, const int* in_sizes, int n_in,
                              void* d_out, int out_size, void* d_ws, size_t ws_size,
                              hipStream_t stream) {
    const float* distance = (const float*)d_in[0];   // [R,128]
    const float* color    = (const float*)d_in[1];   // [R,128,3]
    const float* depth    = (const float*)d_in[2];   // [R,128]
    float* out = (float*)d_out;                      // [R*3] out_color ++ [R*128*3] geometry

    const int R = in_sizes[0] / 128;

    // zero the geometry region (starts at out + R*3 floats; R*12 bytes is 16B-aligned)
    float* geo = out + (size_t)R * 3;
    size_t n4  = (size_t)R * 96;                     // R*128*3/4 float4 elements
    zero_geometry<<<8192, 256, 0, stream>>>((float4*)geo, n4);

    // one wave per ray, 8 rays per 256-thread block
    render_rays<<<(R + 7) / 8, 256, 0, stream>>>(distance, color, depth, out, R);
}